// ShowerGNN_41016937677351
// MI455X (gfx1250) — compile-verified
//
#include <hip/hip_runtime.h>

#define BB 4
#define NN 4096
#define KNB 8
#define EPSV 1e-3f

typedef __attribute__((ext_vector_type(2))) float v2f;
typedef __attribute__((ext_vector_type(8))) float v8f;

__device__ __forceinline__ v8f wmma4(v2f a, v2f b, v8f c) {
  // D(16x16,f32) = A(16x4,f32) x B(4x16,f32) + C
  return __builtin_amdgcn_wmma_f32_16x16x4_f32(
      /*neg_a=*/false, a, /*neg_b=*/false, b,
      /*c_mod=*/(short)0, c, /*reuse_a=*/false, /*reuse_b=*/false);
}

// ---------------- Kernel 1: BatchNorm over F=6, split pos(2)/feat(4) ----------------
__global__ void k_bn1(const float* __restrict__ x,
                      const float* __restrict__ gamma, const float* __restrict__ beta,
                      const float* __restrict__ mean, const float* __restrict__ var,
                      float* __restrict__ pos, float* __restrict__ feat) {
  int i = blockIdx.x * blockDim.x + threadIdx.x;
  if (i >= BB * NN) return;
#pragma unroll
  for (int f = 0; f < 6; ++f) {
    float v = (x[i * 6 + f] - mean[f]) * rsqrtf(var[f] + EPSV) * gamma[f] + beta[f];
    if (f < 2) pos[i * 2 + f] = v;
    else       feat[i * 4 + (f - 2)] = v;
  }
}

// ---------------- Kernel 2: brute-force 2-D KNN (top-8 excluding self) ----------------
__global__ void k_knn(const float* __restrict__ pos, int* __restrict__ knn) {
  const int TB = 256;
  __shared__ float2 tile[TB];
  int b = blockIdx.x / (NN / TB);
  int i = (blockIdx.x % (NN / TB)) * TB + threadIdx.x;   // query index within batch
  const float2* p = (const float2*)(pos + (size_t)b * NN * 2);
  float2 q = p[i];
  float bd[KNB];
  int bi[KNB];
#pragma unroll
  for (int s = 0; s < KNB; ++s) { bd[s] = 3.0e38f; bi[s] = 0; }
  for (int t0 = 0; t0 < NN; t0 += TB) {
    __syncthreads();
    tile[threadIdx.x] = p[t0 + threadIdx.x];
    __syncthreads();
    for (int j = 0; j < TB; ++j) {
      float dx = q.x - tile[j].x;
      float dy = q.y - tile[j].y;
      float d2 = dx * dx + dy * dy;
      int cj = t0 + j;
      if (d2 < bd[KNB - 1] && cj != i) {
        bd[KNB - 1] = d2; bi[KNB - 1] = cj;
#pragma unroll
        for (int s = KNB - 1; s > 0; --s) {
          if (bd[s] < bd[s - 1]) {
            float td = bd[s]; bd[s] = bd[s - 1]; bd[s - 1] = td;
            int ti = bi[s]; bi[s] = bi[s - 1]; bi[s - 1] = ti;
          }
        }
      }
    }
  }
#pragma unroll
  for (int s = 0; s < KNB; ++s) knn[((size_t)b * NN + i) * KNB + s] = bi[s];
}

// ------------- Kernel 3/4: GravNet block (gather-mean + 2-layer MLP via WMMA) -------------
// One wave (32 threads) per tile of 16 points. FD = input feature dim (4 or 32).
template <int FD>
__global__ void __launch_bounds__(32)
k_gravnet(const float* __restrict__ feat, const int* __restrict__ knn,
          const float* __restrict__ w1, const float* __restrict__ b1,   // FD x 32
          const float* __restrict__ w2, const float* __restrict__ b2,   // 32 x 32
          float* __restrict__ out) {                                    // [B*N, 32]
  __shared__ float agg[16][FD];
  __shared__ float hh[16][32];
  const int lane = threadIdx.x;
  const int base = blockIdx.x * 16;        // flat point index base in [0, B*N)
  const int b = base / NN;

  // gather K=8 neighbors and mean
  for (int t = lane; t < 16 * FD; t += 32) {
    int p = t / FD, f = t % FD;
    const int* ki = knn + (size_t)(base + p) * KNB;
    float s = 0.f;
#pragma unroll
    for (int k = 0; k < KNB; ++k)
      s += feat[((size_t)b * NN + ki[k]) * FD + f];
    agg[p][f] = s * (1.f / KNB);
  }
  __syncthreads();

  const int row = lane & 15;            // A-matrix row (M)
  const int kb  = (lane >> 4) * 2;      // K-pair base per ISA 16x4 f32 layout
  const int mb  = (lane >> 4) * 8;      // C-matrix M base per ISA layout

  // layer 1: relu(agg @ w1 + b1) -> hh
  for (int nt = 0; nt < 2; ++nt) {
    int col = nt * 16 + (lane & 15);
    v8f acc = {};
#pragma unroll
    for (int kc = 0; kc < FD / 4; ++kc) {
      int k0 = kc * 4 + kb;
      v2f a = { agg[row][k0], agg[row][k0 + 1] };
      v2f w = { w1[k0 * 32 + col], w1[(k0 + 1) * 32 + col] };
      acc = wmma4(a, w, acc);
    }
    float bias = b1[col];
#pragma unroll
    for (int v = 0; v < 8; ++v)
      hh[mb + v][col] = fmaxf(acc[v] + bias, 0.f);
  }
  __syncthreads();

  // layer 2: hh @ w2 + b2 -> out (no relu)
  for (int nt = 0; nt < 2; ++nt) {
    int col = nt * 16 + (lane & 15);
    v8f acc = {};
#pragma unroll
    for (int kc = 0; kc < 8; ++kc) {
      int k0 = kc * 4 + kb;
      v2f a = { hh[row][k0], hh[row][k0 + 1] };
      v2f w = { w2[k0 * 32 + col], w2[(k0 + 1) * 32 + col] };
      acc = wmma4(a, w, acc);
    }
    float bias = b2[col];
#pragma unroll
    for (int v = 0; v < 8; ++v)
      out[(size_t)(base + mb + v) * 32 + col] = acc[v] + bias;
  }
}

// ------------- Kernel 5: fused dense 68->128->64->32 -> BN2 -> 1 -------------
__global__ void __launch_bounds__(32)
k_dense(const float* __restrict__ pos, const float* __restrict__ h1,
        const float* __restrict__ h2,
        const float* __restrict__ d1w, const float* __restrict__ d1b,
        const float* __restrict__ d2w, const float* __restrict__ d2b,
        const float* __restrict__ d3w, const float* __restrict__ d3b,
        const float* __restrict__ g2, const float* __restrict__ be2,
        const float* __restrict__ mn2, const float* __restrict__ vr2,
        const float* __restrict__ ow, const float* __restrict__ ob,
        float* __restrict__ out) {
  __shared__ float xx[16][68];
  __shared__ float y1[16][128];
  __shared__ float y2[16][64];
  __shared__ float y3[16][32];
  const int lane = threadIdx.x;
  const int base = blockIdx.x * 16;

  // build x = [pos, h1, pos, h2] (68)
  for (int t = lane; t < 16 * 68; t += 32) {
    int p = t / 68, c = t % 68;
    float v;
    if (c < 2)       v = pos[(size_t)(base + p) * 2 + c];
    else if (c < 34) v = h1[(size_t)(base + p) * 32 + (c - 2)];
    else if (c < 36) v = pos[(size_t)(base + p) * 2 + (c - 34)];
    else             v = h2[(size_t)(base + p) * 32 + (c - 36)];
    xx[p][c] = v;
  }
  __syncthreads();

  const int row = lane & 15;
  const int kb  = (lane >> 4) * 2;
  const int mb  = (lane >> 4) * 8;

  // d1: 68 -> 128, relu
  for (int nt = 0; nt < 8; ++nt) {
    int col = nt * 16 + (lane & 15);
    v8f acc = {};
#pragma unroll
    for (int kc = 0; kc < 17; ++kc) {
      int k0 = kc * 4 + kb;
      v2f a = { xx[row][k0], xx[row][k0 + 1] };
      v2f w = { d1w[k0 * 128 + col], d1w[(k0 + 1) * 128 + col] };
      acc = wmma4(a, w, acc);
    }
    float bias = d1b[col];
#pragma unroll
    for (int v = 0; v < 8; ++v) y1[mb + v][col] = fmaxf(acc[v] + bias, 0.f);
  }
  __syncthreads();

  // d2: 128 -> 64, relu
  for (int nt = 0; nt < 4; ++nt) {
    int col = nt * 16 + (lane & 15);
    v8f acc = {};
#pragma unroll 8
    for (int kc = 0; kc < 32; ++kc) {
      int k0 = kc * 4 + kb;
      v2f a = { y1[row][k0], y1[row][k0 + 1] };
      v2f w = { d2w[k0 * 64 + col], d2w[(k0 + 1) * 64 + col] };
      acc = wmma4(a, w, acc);
    }
    float bias = d2b[col];
#pragma unroll
    for (int v = 0; v < 8; ++v) y2[mb + v][col] = fmaxf(acc[v] + bias, 0.f);
  }
  __syncthreads();

  // d3: 64 -> 32, relu, then BN2
  for (int nt = 0; nt < 2; ++nt) {
    int col = nt * 16 + (lane & 15);
    v8f acc = {};
#pragma unroll
    for (int kc = 0; kc < 16; ++kc) {
      int k0 = kc * 4 + kb;
      v2f a = { y2[row][k0], y2[row][k0 + 1] };
      v2f w = { d3w[k0 * 32 + col], d3w[(k0 + 1) * 32 + col] };
      acc = wmma4(a, w, acc);
    }
    float bias = d3b[col];
    float sc = rsqrtf(vr2[col] + EPSV) * g2[col];
    float mu = mn2[col], bt = be2[col];
#pragma unroll
    for (int v = 0; v < 8; ++v) {
      float r = fmaxf(acc[v] + bias, 0.f);
      y3[mb + v][col] = (r - mu) * sc + bt;
    }
  }
  __syncthreads();

  // out: 32 -> 1 per point
  if (lane < 16) {
    float s = ob[0];
#pragma unroll
    for (int c = 0; c < 32; ++c) s += y3[lane][c] * ow[c];
    out[base + lane] = s;
  }
}

extern "C" void kernel_launch(void* const* d_in, const int* in_sizes, int n_in,
                              void* d_out, int out_size, void* d_ws, size_t ws_size,
                              hipStream_t stream) {
  (void)in_sizes; (void)n_in; (void)out_size; (void)ws_size;
  const float* inputs = (const float*)d_in[0];
  const float* bn1g = (const float*)d_in[1];
  const float* bn1b = (const float*)d_in[2];
  const float* bn1m = (const float*)d_in[3];
  const float* bn1v = (const float*)d_in[4];
  const float* g1w1 = (const float*)d_in[5];
  const float* g1b1 = (const float*)d_in[6];
  const float* g1w2 = (const float*)d_in[7];
  const float* g1b2 = (const float*)d_in[8];
  const float* g2w1 = (const float*)d_in[9];
  const float* g2b1 = (const float*)d_in[10];
  const float* g2w2 = (const float*)d_in[11];
  const float* g2b2 = (const float*)d_in[12];
  const float* d1w  = (const float*)d_in[13];
  const float* d1b  = (const float*)d_in[14];
  const float* d2w  = (const float*)d_in[15];
  const float* d2b  = (const float*)d_in[16];
  const float* d3w  = (const float*)d_in[17];
  const float* d3b  = (const float*)d_in[18];
  const float* bn2g = (const float*)d_in[19];
  const float* bn2b = (const float*)d_in[20];
  const float* bn2m = (const float*)d_in[21];
  const float* bn2v = (const float*)d_in[22];
  const float* outw = (const float*)d_in[23];
  const float* outb = (const float*)d_in[24];

  float* pos   = (float*)d_ws;               // B*N*2
  float* feat0 = pos + BB * NN * 2;          // B*N*4
  float* h1    = feat0 + BB * NN * 4;        // B*N*32
  float* h2    = h1 + BB * NN * 32;          // B*N*32
  int*   knn   = (int*)(h2 + BB * NN * 32);  // B*N*8

  k_bn1<<<(BB * NN + 255) / 256, 256, 0, stream>>>(inputs, bn1g, bn1b, bn1m, bn1v, pos, feat0);
  k_knn<<<BB * (NN / 256), 256, 0, stream>>>(pos, knn);
  k_gravnet<4><<<BB * NN / 16, 32, 0, stream>>>(feat0, knn, g1w1, g1b1, g1w2, g1b2, h1);
  k_gravnet<32><<<BB * NN / 16, 32, 0, stream>>>(h1, knn, g2w1, g2b1, g2w2, g2b2, h2);
  k_dense<<<BB * NN / 16, 32, 0, stream>>>(pos, h1, h2, d1w, d1b, d2w, d2b, d3w, d3b,
                                           bn2g, bn2b, bn2m, bn2v, outw, outb,
                                           (float*)d_out);
}